// DampingGCN_83691732730293
// MI455X (gfx1250) — compile-verified
//
#include <hip/hip_runtime.h>

typedef __attribute__((ext_vector_type(2))) float v2f;
typedef __attribute__((ext_vector_type(4))) float v4f;
typedef __attribute__((ext_vector_type(8))) float v8f;

#define HD 64

// ---------------- GCN normalization ----------------

__global__ __launch_bounds__(256) void k_deg(const int* __restrict__ ei,
                                             unsigned* __restrict__ deg, int E) {
  int e = blockIdx.x * 256 + threadIdx.x;
  if (e < E) atomicAdd(&deg[ei[E + e]], 1u);   // dst-degree
}

__global__ __launch_bounds__(256) void k_dis(const unsigned* __restrict__ deg,
                                             float* __restrict__ dis, int N) {
  int i = blockIdx.x * 256 + threadIdx.x;
  if (i < N) dis[i] = rsqrtf((float)(deg[i] + 1u));  // +1 self loop; deg>=1 always
}

__global__ __launch_bounds__(256) void k_norm(const int* __restrict__ ei,
                                              const float* __restrict__ dis,
                                              float* __restrict__ nrm, int E) {
  int e = blockIdx.x * 256 + threadIdx.x;
  if (e < E) nrm[e] = dis[ei[e]] * dis[ei[E + e]];
}

// ---------------- Layer 1 dense (2 -> 64), vectorized b128 ----------------

__global__ __launch_bounds__(256) void k_lin1(const float* __restrict__ x,
                                              const float* __restrict__ W1,
                                              float* __restrict__ out, int N) {
  int t = blockIdx.x * 256 + threadIdx.x;           // one float4 per thread
  if (t >= N * (HD / 4)) return;
  int n = t >> 4, q = t & 15;
  v4f w0 = *(const v4f*)(W1 + q * 4);
  v4f w1 = *(const v4f*)(W1 + HD + q * 4);
  float x0 = x[2 * n], x1 = x[2 * n + 1];
  *(v4f*)(out + (size_t)n * HD + q * 4) = w0 * x0 + w1 * x1;
}

// ---------------- Dense 64x64 transform via fp32 WMMA ----------------
// Out[N x 64] = relu(A + bias) @ W.  One wave per 16x16 tile, 16 chained
// V_WMMA_F32_16X16X4_F32 over K=64.  W and bias staged in LDS; the previous
// layer's bias+ReLU is fused into the A-fragment load (saves a full h pass).

__global__ __launch_bounds__(256) void k_gemm64(const float* __restrict__ A,
                                                const float* __restrict__ W,
                                                const float* __restrict__ bias,
                                                float* __restrict__ Out, int N) {
  __shared__ float sW[HD * HD];
  __shared__ float sB[HD];
  for (int i = threadIdx.x; i < HD * HD; i += 256) sW[i] = W[i];
  if (threadIdx.x < HD) sB[threadIdx.x] = bias[threadIdx.x];
  __syncthreads();

  const int wave = threadIdx.x >> 5;
  const int lane = threadIdx.x & 31;
  const int tile = blockIdx.x * 8 + wave;
  const int rowTiles = (N + 15) >> 4;
  if (tile >= rowTiles * 4) return;          // uniform per wave -> EXEC stays all-1s
  const int m0 = (tile >> 2) << 4;
  const int n0 = (tile & 3) << 4;
  const int half = lane >> 4;                // 0: lanes 0-15, 1: lanes 16-31
  const int l16 = lane & 15;

  // A fragments: 32-bit A 16x4 layout — lane<16 holds {K+0,K+1}, lane>=16 {K+2,K+3}
  int ar = m0 + l16; if (ar >= N) ar = N - 1;            // clamp; stores guarded below
  const float* arow = A + (size_t)ar * HD + half * 2;
  v2f afrag[16];
#pragma unroll
  for (int k = 0; k < 16; ++k) {
    v2f a = *(const v2f*)(arow + 4 * k);
    const int c0 = 4 * k + half * 2;                     // fused bias + ReLU
    a.x = fmaxf(a.x + sB[c0], 0.f);
    a.y = fmaxf(a.y + sB[c0 + 1], 0.f);
    afrag[k] = a;
  }

  // B fragments: 4x16 tile, row K striped across lanes (N=lane), mirrored halves
  v2f bfrag[16];
#pragma unroll
  for (int k = 0; k < 16; ++k) {
    const int r = 4 * k + half * 2;
    v2f b;
    b.x = sW[r * HD + n0 + l16];
    b.y = sW[(r + 1) * HD + n0 + l16];
    bfrag[k] = b;
  }

  v8f c = {};
#pragma unroll
  for (int k = 0; k < 16; ++k)
    c = __builtin_amdgcn_wmma_f32_16x16x4_f32(false, afrag[k], false, bfrag[k],
                                              (short)0, c, false, false);

  // C/D layout: VGPR j -> row m0 + j + half*8, col n0 + l16
  const int rbase = m0 + half * 8;
#pragma unroll
  for (int j = 0; j < 8; ++j) {
    const int row = rbase + j;
    if (row < N) Out[(size_t)row * HD + n0 + l16] = c[j];
  }
}

// ---------------- Aggregation ----------------

// Self-loop contribution initializes the output (full overwrite, no zeroing), b128
__global__ __launch_bounds__(256) void k_selfagg(const float* __restrict__ tmp,
                                                 const float* __restrict__ dis,
                                                 float* __restrict__ out, int N) {
  int t = blockIdx.x * 256 + threadIdx.x;           // one float4 per thread
  if (t >= N * (HD / 4)) return;
  float d = dis[t >> 4];
  v4f v = *(const v4f*)(tmp + (size_t)t * 4) * (d * d);
  *(v4f*)(out + (size_t)t * 4) = v;
}

// 16 threads per edge: b128 gather, scale, 4x hardware global_atomic_add_f32
__global__ __launch_bounds__(256) void k_edgeagg(const int* __restrict__ ei,
                                                 const float* __restrict__ nrm,
                                                 const float* __restrict__ tmp,
                                                 float* __restrict__ out, int E) {
  long long t = (long long)blockIdx.x * 256 + threadIdx.x;
  if (t >= (long long)E * (HD / 4)) return;
  int e = (int)(t >> 4), q = (int)(t & 15);
  int s = ei[e], d = ei[E + e];
  float w = nrm[e];
  v4f m = *(const v4f*)(tmp + (size_t)s * HD + q * 4) * w;
  float* o = out + (size_t)d * HD + q * 4;
  unsafeAtomicAdd(o + 0, m.x);
  unsafeAtomicAdd(o + 1, m.y);
  unsafeAtomicAdd(o + 2, m.z);
  unsafeAtomicAdd(o + 3, m.w);
}

// ---------------- Final projection relu(h+b3) @ Wl + bl, one wave per node ----

__global__ __launch_bounds__(256) void k_final(const float* __restrict__ h,
                                               const float* __restrict__ b3,
                                               const float* __restrict__ Wl,
                                               const float* __restrict__ bl,
                                               float* __restrict__ y, int N) {
  int gt = blockIdx.x * 256 + threadIdx.x;
  int w = gt >> 5, lane = gt & 31;
  if (w >= N) return;
  const float* row = h + (size_t)w * HD;
  float a0 = fmaxf(row[lane] + b3[lane], 0.f);
  float a1 = fmaxf(row[lane + 32] + b3[lane + 32], 0.f);
  float p = fmaf(a0, Wl[lane], a1 * Wl[lane + 32]);
#pragma unroll
  for (int off = 16; off > 0; off >>= 1) p += __shfl_xor(p, off, 32);
  if (lane == 0) y[w] = p + bl[0];
}

// ---------------- Launch ----------------

extern "C" void kernel_launch(void* const* d_in, const int* in_sizes, int n_in,
                              void* d_out, int out_size, void* d_ws, size_t ws_size,
                              hipStream_t stream) {
  (void)n_in; (void)out_size; (void)ws_size;
  const float* x  = (const float*)d_in[0];
  const int*   ei = (const int*)d_in[1];
  const float* W1 = (const float*)d_in[2];
  const float* b1 = (const float*)d_in[3];
  const float* W2 = (const float*)d_in[4];
  const float* b2 = (const float*)d_in[5];
  const float* W3 = (const float*)d_in[6];
  const float* b3 = (const float*)d_in[7];
  const float* Wl = (const float*)d_in[8];
  const float* bl = (const float*)d_in[9];
  const int N = in_sizes[0] / 2;
  const int E = in_sizes[1] / 2;

  char* p = (char*)d_ws;
  auto alloc = [&](size_t bytes) {
    char* r = p; p += (bytes + 255) & ~(size_t)255; return r;
  };
  unsigned* deg = (unsigned*)alloc((size_t)N * 4);
  float* dis  = (float*)alloc((size_t)N * 4);
  float* nrm  = (float*)alloc((size_t)E * 4);
  float* bufA = (float*)alloc((size_t)N * HD * 4);
  float* bufB = (float*)alloc((size_t)N * HD * 4);

  auto nb = [](long long n) { return (unsigned)((n + 255) / 256); };
  const int NT = 256;

  hipMemsetAsync(deg, 0, (size_t)N * 4, stream);   // graph-capture safe
  k_deg <<<nb(E), NT, 0, stream>>>(ei, deg, E);
  k_dis <<<nb(N), NT, 0, stream>>>(deg, dis, N);
  k_norm<<<nb(E), NT, 0, stream>>>(ei, dis, nrm, E);

  const long long nv = (long long)N * (HD / 4);    // float4 work-items per h pass
  const long long ev = (long long)E * (HD / 4);

  // Layer 1: dense 2->64, then aggregate (bias+ReLU deferred into consumers)
  k_lin1   <<<nb(nv), NT, 0, stream>>>(x, W1, bufA, N);
  k_selfagg<<<nb(nv), NT, 0, stream>>>(bufA, dis, bufB, N);
  k_edgeagg<<<nb(ev), NT, 0, stream>>>(ei, nrm, bufA, bufB, E);

  // Layers 2 and 3: fused relu(h + b_prev) @ W via WMMA, then aggregate
  const unsigned gemmBlocks = (unsigned)((((N + 15) / 16) * 4 + 7) / 8);
  k_gemm64 <<<gemmBlocks, NT, 0, stream>>>(bufB, W2, b1, bufA, N);
  k_selfagg<<<nb(nv), NT, 0, stream>>>(bufA, dis, bufB, N);
  k_edgeagg<<<nb(ev), NT, 0, stream>>>(ei, nrm, bufA, bufB, E);

  k_gemm64 <<<gemmBlocks, NT, 0, stream>>>(bufB, W3, b2, bufA, N);
  k_selfagg<<<nb(nv), NT, 0, stream>>>(bufA, dis, bufB, N);
  k_edgeagg<<<nb(ev), NT, 0, stream>>>(ei, nrm, bufA, bufB, E);

  // Final: relu(h + b3) @ Wl + bl
  k_final<<<nb((long long)N * 32), NT, 0, stream>>>(bufB, b3, Wl, bl, (float*)d_out, N);
}